// DAGR_120259084646
// MI455X (gfx1250) — compile-verified
//
#include <hip/hip_runtime.h>
#include <stdint.h>
#include <stddef.h>

#define BATCH   16
#define NANCH   2048
#define NCLS    80
#define ROWLEN  85          // 5 + C floats per anchor
#define NTHR    1024        // 32 wave32 waves; each thread owns 2 anchors
#define CONF_T  0.25f
#define IOU_T   0.45f
#define MINSZ   5.0f

// ---- gfx1250 async global->LDS helpers -------------------------------------
// low 32 bits of a flat shared-aperture address == LDS byte offset
__device__ __forceinline__ uint32_t lds_off(const void* p) {
    return (uint32_t)(uintptr_t)p;
}
__device__ __forceinline__ void async_g2l_b32(uint32_t dst_lds, const void* gsrc) {
    asm volatile("global_load_async_to_lds_b32 %0, %1, off"
                 :: "v"(dst_lds), "v"((uint64_t)(uintptr_t)gsrc)
                 : "memory");
}
__device__ __forceinline__ void wait_async0() {
    asm volatile("s_wait_asynccnt 0" ::: "memory");
}
// ----------------------------------------------------------------------------

// meta word layout: [11:0] anchor idx | [19:12] class | bit20 pos_wh |
//                   bit21 valid_xyxy | bit22 conf | bit23 valid
#define M_PWH   (1u << 20)
#define M_VXY   (1u << 21)
#define M_CONF  (1u << 22)
#define M_VALID (1u << 23)

__global__ __launch_bounds__(NTHR)
void DAGR_nms_kernel(const float* __restrict__ pred, float* __restrict__ out) {
    __shared__ float    sBox[NANCH][4];   // 32 KB: coords, later final boxes (sorted)
    __shared__ float    sKey[NANCH];      //  8 KB: score, later sort key
    __shared__ uint32_t sMeta[NANCH];     //  8 KB
    __shared__ int      sKeep[NANCH];     //  8 KB: one-writer keep flags
    __shared__ int      sCnt[3];          // conf count, xywh votes, xyxy votes

    const int tid = threadIdx.x;
    const int b   = blockIdx.x;
    const float* img = pred + (size_t)b * NANCH * ROWLEN;

    if (tid < 3) sCnt[tid] = 0;

    // ---- phase 0: async-stage coords + objectness into LDS ----
    for (int r = 0; r < 2; ++r) {
        int n = tid + r * NTHR;
        const float* rowp = img + (size_t)n * ROWLEN;
        async_g2l_b32(lds_off(&sBox[n][0]), rowp + 0);
        async_g2l_b32(lds_off(&sBox[n][1]), rowp + 1);
        async_g2l_b32(lds_off(&sBox[n][2]), rowp + 2);
        async_g2l_b32(lds_off(&sBox[n][3]), rowp + 3);
        async_g2l_b32(lds_off(&sKey[n]),    rowp + 4);
    }
    wait_async0();
    __syncthreads();

    // ---- phase 1: class argmax + flags ----
    for (int r = 0; r < 2; ++r) {
        int n = tid + r * NTHR;
        const float* crow = img + (size_t)n * ROWLEN + 5;
        __builtin_prefetch(crow, 0, 2);
        float best = crow[0]; int bc = 0;
        #pragma unroll 4
        for (int c = 1; c < NCLS; ++c) {
            float v = crow[c];
            if (v > best) { best = v; bc = c; }   // strict > == first-max (jnp.argmax)
        }
        float x = sBox[n][0], y = sBox[n][1], w = sBox[n][2], h = sBox[n][3];
        float sc = sKey[n];
        uint32_t m = (uint32_t)n | ((uint32_t)bc << 12);
        if ((w > 0.f) && (h > 0.f)) m |= M_PWH;
        if ((w > x)   && (h > y))   m |= M_VXY;
        if (sc >= CONF_T) { m |= M_CONF; atomicAdd(&sCnt[0], 1); }
        sMeta[n] = m;
    }
    __syncthreads();

    // ---- phase 2: top-5 fallback if no anchor passed threshold (rare) ----
    if (tid == 0 && sCnt[0] == 0) {
        for (int it = 0; it < 5; ++it) {
            float bk = -1.0e30f; int bi = 0;
            for (int n = 0; n < NANCH; ++n) {
                if (sMeta[n] & M_CONF) continue;
                float v = sKey[n];
                if (v > bk) { bk = v; bi = n; }   // ties -> lowest index
            }
            sMeta[bi] |= M_CONF;
        }
    }
    __syncthreads();

    // ---- phase 3: format vote ----
    for (int r = 0; r < 2; ++r) {
        int n = tid + r * NTHR;
        uint32_t m = sMeta[n];
        if (m & M_CONF) {
            if (m & M_PWH) atomicAdd(&sCnt[1], 1);
            if (m & M_VXY) atomicAdd(&sCnt[2], 1);
        }
    }
    __syncthreads();
    const bool useXYWH = (sCnt[1] >= sCnt[2]);

    // ---- phase 4: finalize boxes + sort key ----
    for (int r = 0; r < 2; ++r) {
        int n = tid + r * NTHR;
        uint32_t m = sMeta[n];
        float x = sBox[n][0], y = sBox[n][1], w = sBox[n][2], h = sBox[n][3];
        float sc = sKey[n];
        float b0, b1, b2, b3; bool valid;
        if (useXYWH) {
            float W = fmaxf(w, MINSZ), H = fmaxf(h, MINSZ);
            b0 = x - W * 0.5f; b1 = y - H * 0.5f;
            b2 = x + W * 0.5f; b3 = y + H * 0.5f;
            float area = (b2 - b0) * (b3 - b1);
            bool cv = (b2 > b0) && (b3 > b1) && (area >= MINSZ * MINSZ);
            valid = (m & M_CONF) && (m & M_PWH) && cv;
        } else {
            b0 = x; b1 = y; b2 = w; b3 = h;
            valid = (m & M_CONF) && (m & M_VXY);
        }
        sBox[n][0] = b0; sBox[n][1] = b1; sBox[n][2] = b2; sBox[n][3] = b3;
        sKey[n] = valid ? sc : -1.0e9f;
        sMeta[n] = valid ? (m | M_VALID) : (m & ~M_VALID);
    }
    __syncthreads();

    // ---- phase 5: bitonic sort of (key desc, anchor asc); permutes sKey/sMeta ----
    for (int k = 2; k <= NANCH; k <<= 1) {
        for (int j = k >> 1; j > 0; j >>= 1) {
            for (int r = 0; r < 2; ++r) {
                int i = tid + r * NTHR;
                int ixj = i ^ j;
                if (ixj > i) {
                    float ka = sKey[i], kb = sKey[ixj];
                    uint32_t ma = sMeta[i], mb = sMeta[ixj];
                    // "a belongs after b" in descending order (strict total order)
                    bool aAfter = (ka < kb) ||
                                  ((ka == kb) && ((ma & 0xFFFu) > (mb & 0xFFFu)));
                    bool dirDesc = ((i & k) == 0);
                    if (aAfter == dirDesc) {
                        sKey[i] = kb;  sKey[ixj] = ka;
                        sMeta[i] = mb; sMeta[ixj] = ma;
                    }
                }
            }
            __syncthreads();
        }
    }

    // ---- phase 6: gather boxes into sorted order; init keep ----
    float gb[2][4];
    for (int r = 0; r < 2; ++r) {
        int p = tid + r * NTHR;
        int o = sMeta[p] & 0xFFFu;
        gb[r][0] = sBox[o][0]; gb[r][1] = sBox[o][1];
        gb[r][2] = sBox[o][2]; gb[r][3] = sBox[o][3];
    }
    __syncthreads();
    for (int r = 0; r < 2; ++r) {
        int p = tid + r * NTHR;
        sBox[p][0] = gb[r][0]; sBox[p][1] = gb[r][1];
        sBox[p][2] = gb[r][2]; sBox[p][3] = gb[r][3];
        sKeep[p] = (sMeta[p] >> 23) & 1;
    }
    __syncthreads();

    // ---- phase 7: greedy NMS (class equality == exact class-offset trick) ----
    float jx1[2], jy1[2], jx2[2], jy2[2], jar[2];
    int   jcls[2], jkeep[2];
    for (int r = 0; r < 2; ++r) {
        int p = tid + r * NTHR;
        jx1[r] = sBox[p][0]; jy1[r] = sBox[p][1];
        jx2[r] = sBox[p][2]; jy2[r] = sBox[p][3];
        jar[r] = fmaxf(jx2[r] - jx1[r], 0.f) * fmaxf(jy2[r] - jy1[r], 0.f);
        jcls[r] = (int)((sMeta[p] >> 12) & 0xFFu);
        jkeep[r] = sKeep[p];
    }
    for (int i = 0; i < NANCH; ++i) {
        __syncthreads();                       // publish previous suppressions
        if (sKeep[i]) {                        // uniform branch (LDS broadcast)
            float ax1 = sBox[i][0], ay1 = sBox[i][1];
            float ax2 = sBox[i][2], ay2 = sBox[i][3];
            int   acls = (int)((sMeta[i] >> 12) & 0xFFu);
            float aar = fmaxf(ax2 - ax1, 0.f) * fmaxf(ay2 - ay1, 0.f);
            for (int r = 0; r < 2; ++r) {
                int jp = tid + r * NTHR;
                if (jp > i && jkeep[r] && jcls[r] == acls) {
                    float ix = fmaxf(fminf(ax2, jx2[r]) - fmaxf(ax1, jx1[r]), 0.f);
                    float iy = fmaxf(fminf(ay2, jy2[r]) - fmaxf(ay1, jy1[r]), 0.f);
                    float inter = ix * iy;
                    float uni = aar + jar[r] - inter;
                    float iou = inter / fmaxf(uni, 1e-9f);
                    if (iou > IOU_T) { jkeep[r] = 0; sKeep[jp] = 0; }
                }
            }
        }
    }
    __syncthreads();

    // ---- phase 8: scatter output rows back to original anchor order ----
    for (int r = 0; r < 2; ++r) {
        int p = tid + r * NTHR;
        uint32_t m = sMeta[p];
        int o = m & 0xFFFu;
        int kept = sKeep[p];
        float kf = kept ? 1.f : 0.f;
        float* orow = out + ((size_t)b * NANCH + (size_t)o) * 6;
        orow[0] = sBox[p][0] * kf;
        orow[1] = sBox[p][1] * kf;
        orow[2] = sBox[p][2] * kf;
        orow[3] = sBox[p][3] * kf;
        orow[4] = kept ? sKey[p] : 0.0f;                       // key==score when kept
        orow[5] = (float)((m >> 12) & 0xFFu) * kf;
    }
}

extern "C" void kernel_launch(void* const* d_in, const int* in_sizes, int n_in,
                              void* d_out, int out_size, void* d_ws, size_t ws_size,
                              hipStream_t stream) {
    (void)in_sizes; (void)n_in; (void)out_size; (void)d_ws; (void)ws_size;
    const float* pred = (const float*)d_in[0];
    float* out = (float*)d_out;
    DAGR_nms_kernel<<<BATCH, NTHR, 0, stream>>>(pred, out);
}